// GINConcat_45380624450159
// MI455X (gfx1250) — compile-verified
//
#include <hip/hip_runtime.h>
#include <hip/hip_bf16.h>

#define NODES   200000
#define EDGES   3200000
#define GRAPHS  1024
#define FEAT    32
#define NELEM   (NODES * FEAT)      // 6,400,000
#define HCATC   128                 // 3*32 + 32

typedef float v2f __attribute__((ext_vector_type(2)));
typedef float v8f __attribute__((ext_vector_type(8)));

// ---------------------------------------------------------------- utilities
__global__ void k_zero(float* __restrict__ p, int n) {
    int i = blockIdx.x * blockDim.x + threadIdx.x;
    if (i < n) p[i] = 0.0f;
}

// ------------------------------------------------------- edge scatter-add
// wave per edge, lane per feature: agg[dst] += x[src]
// lanes of a wave touch one contiguous 128B row for both the gather and
// the scatter-add, so the f32 atomics coalesce into full-line L2 ops.
__global__ void k_edge_agg(const float* __restrict__ x,
                           const int* __restrict__ src,
                           const int* __restrict__ dst,
                           float* __restrict__ agg) {
    int tid = blockIdx.x * blockDim.x + threadIdx.x;
    int e = tid >> 5;
    int f = tid & 31;
    if (e < EDGES) {
        int s = src[e];
        int d = dst[e];
        atomicAdd(&agg[d * FEAT + f], x[s * FEAT + f]);
    }
}

// ------------------------------------------------------- graph pooling
__global__ void k_pool(const float* __restrict__ v,
                       const int* __restrict__ batch,
                       float* __restrict__ hcat, int col_off) {
    int i = blockIdx.x * blockDim.x + threadIdx.x;
    if (i < NELEM) {
        int n = i >> 5, c = i & 31;
        atomicAdd(&hcat[batch[n] * HCATC + col_off + c], v[i]);
    }
}

// ----------------------------------------- fused outer BN + ReLU + pool
__global__ void k_bn_relu_pool(const float* __restrict__ h,
                               const float* __restrict__ ss,   // [0:32)=scale [32:64)=shift
                               const int* __restrict__ batch,
                               float* __restrict__ xout,
                               float* __restrict__ hcat, int col_off) {
    int i = blockIdx.x * blockDim.x + threadIdx.x;
    if (i < NELEM) {
        int n = i >> 5, c = i & 31;
        float v = fmaxf(h[i] * ss[c] + ss[32 + c], 0.0f);
        xout[i] = v;
        atomicAdd(&hcat[batch[n] * HCATC + col_off + c], v);
    }
}

__global__ void k_finalize(const float* __restrict__ stats,
                           const float* __restrict__ g,
                           const float* __restrict__ b,
                           float* __restrict__ ss) {
    int c = threadIdx.x;
    if (c < 32) {
        const float invN = 1.0f / (float)NODES;
        float mean = stats[c] * invN;
        float var  = stats[32 + c] * invN - mean * mean;
        float sc   = g[c] * rsqrtf(var + 1e-5f);
        ss[c]      = sc;
        ss[32 + c] = b[c] - mean * sc;
    }
}

// ------------------------------------------------------------- WMMA GEMM
// out[node, 0:32] = pre(node,:) @ W(32x32) + bias,
// fused with per-column sum / sum-of-squares for the following BatchNorm.
//   mode 0: pre = in0 + in1              (GIN: x + agg)
//   mode 1: pre = relu(in0*scale+shift)  (inner BN + ReLU fused)
// block = 256 thr = 8 waves; 128 nodes/block; wave owns a 16-row stripe.
// Uses V_WMMA_F32_16X16X4_F32, 8 K-steps, 2 N-tiles; C routed through LDS
// for coalesced b128 stores and the fused stats reduction.
__global__ __launch_bounds__(256) void k_gemm32(const float* __restrict__ in0,
                                                const float* __restrict__ in1,
                                                const float* __restrict__ ss,
                                                const float* __restrict__ W,
                                                const float* __restrict__ bias,
                                                float* __restrict__ out,
                                                float* __restrict__ gstats,
                                                int mode) {
    __shared__ float lds[128 * FEAT];
    __shared__ float sstats[64];
    const int t = threadIdx.x;
    const int base = blockIdx.x * 128;
    const bool full = (base + 128) <= NODES;   // uniform: only last block partial

    // ---- stage pre-activation tile into LDS (float4 -> ds_store_b128)
    #pragma unroll
    for (int it = 0; it < 4; it++) {
        int off4 = t + it * 256;          // float4 index, 0..1023
        int elem = off4 * 4;              // element index, multiple of 4
        int node = base + (elem >> 5);
        int c = elem & 31;                // column, multiple of 4
        float4 v = make_float4(0.f, 0.f, 0.f, 0.f);
        if (full || node < NODES) {
            float4 a = *(const float4*)(in0 + node * FEAT + c);
            if (mode == 0) {
                float4 b = *(const float4*)(in1 + node * FEAT + c);
                v = make_float4(a.x + b.x, a.y + b.y, a.z + b.z, a.w + b.w);
            } else {
                v.x = fmaxf(a.x * ss[c + 0] + ss[32 + c + 0], 0.f);
                v.y = fmaxf(a.y * ss[c + 1] + ss[32 + c + 1], 0.f);
                v.z = fmaxf(a.z * ss[c + 2] + ss[32 + c + 2], 0.f);
                v.w = fmaxf(a.w * ss[c + 3] + ss[32 + c + 3], 0.f);
            }
        }
        ((float4*)lds)[off4] = v;
    }
    __syncthreads();

    const int wave = t >> 5;
    const int lane = t & 31;
    const int lo16 = lane & 15;
    const int hi   = lane >> 4;           // A/B half: K+{0,1} vs K+{2,3}
    const int lrow = (wave << 4) + lo16;  // A-matrix row within block tile

    // A fragments: ISA 16x4 f32 layout (per lane: two consecutive K values)
    v2f afrag[8];
    #pragma unroll
    for (int k = 0; k < 8; k++) {
        int kb = 4 * k + 2 * hi;
        afrag[k].x = lds[lrow * FEAT + kb];
        afrag[k].y = lds[lrow * FEAT + kb + 1];
    }
    __syncthreads();                      // all A reads done before C overwrites

    if (t < 64) sstats[t] = 0.0f;

    const int n0 = lo16;                  // C column for nt=0
    #pragma unroll
    for (int nt = 0; nt < 2; nt++) {
        v8f acc = {0.f, 0.f, 0.f, 0.f, 0.f, 0.f, 0.f, 0.f};
        int n = nt * 16 + n0;
        #pragma unroll
        for (int k = 0; k < 8; k++) {
            int kb = 4 * k + 2 * hi;
            v2f bfrag;
            bfrag.x = W[kb * 32 + n];
            bfrag.y = W[(kb + 1) * 32 + n];
            acc = __builtin_amdgcn_wmma_f32_16x16x4_f32(
                false, afrag[k], false, bfrag, 0, acc, false, false);
        }
        float bv = bias[n];
        // C layout: lane holds column n; VGPR r -> row r + 8*hi
        #pragma unroll
        for (int r = 0; r < 8; r++) {
            int m = (wave << 4) + r + 8 * hi;
            lds[m * FEAT + n] = acc[r] + bv;
        }
    }
    __syncthreads();

    // ---- coalesced b128 store + fused BN statistics
    const int cst = (t * 4) & 31;         // this thread's column group (const)
    float ps[4] = {0.f, 0.f, 0.f, 0.f};
    float pq[4] = {0.f, 0.f, 0.f, 0.f};
    #pragma unroll
    for (int it = 0; it < 4; it++) {
        int off4 = t + it * 256;
        int elem = off4 * 4;
        int node = base + (elem >> 5);
        if (full || node < NODES) {
            float4 v = ((float4*)lds)[off4];
            *(float4*)(out + node * FEAT + (elem & 31)) = v;
            ps[0] += v.x; pq[0] += v.x * v.x;
            ps[1] += v.y; pq[1] += v.y * v.y;
            ps[2] += v.z; pq[2] += v.z * v.z;
            ps[3] += v.w; pq[3] += v.w * v.w;
        }
    }
    #pragma unroll
    for (int j = 0; j < 4; j++) {
        atomicAdd(&sstats[cst + j], ps[j]);
        atomicAdd(&sstats[32 + cst + j], pq[j]);
    }
    __syncthreads();
    if (t < 64) atomicAdd(&gstats[t], sstats[t]);
}

// ------------------------------------------------------------- readout MLP
__global__ void k_readout(const float* __restrict__ hcat,
                          const float* __restrict__ W0,
                          const float* __restrict__ b0,
                          const float* __restrict__ W1,
                          const float* __restrict__ b1,
                          float* __restrict__ out) {
    int g = blockIdx.x * blockDim.x + threadIdx.x;
    if (g >= GRAPHS) return;
    const float* row = hcat + g * HCATC;
    float o = 0.0f;
    for (int j = 0; j < 32; j++) {
        float acc = b0[j];
        for (int k = 0; k < HCATC; k++) acc += row[k] * W0[k * 32 + j];
        o += fmaxf(acc, 0.0f) * W1[j];
    }
    out[g] = o + b1[0];
}

// ---------------------------------------------------------------- launcher
extern "C" void kernel_launch(void* const* d_in, const int* in_sizes, int n_in,
                              void* d_out, int out_size, void* d_ws, size_t ws_size,
                              hipStream_t stream) {
    const float* x      = (const float*)d_in[0];
    const int*   ei     = (const int*)d_in[1];      // [2, EDGES]
    const int*   batch  = (const int*)d_in[2];
    const float* W1s    = (const float*)d_in[3];    // [3,32,32]
    const float* b1s    = (const float*)d_in[4];    // [3,32]
    const float* bng    = (const float*)d_in[5];
    const float* bnb    = (const float*)d_in[6];
    const float* W2s    = (const float*)d_in[7];
    const float* b2s    = (const float*)d_in[8];
    const float* obng   = (const float*)d_in[9];
    const float* obnb   = (const float*)d_in[10];
    const float* lin0W  = (const float*)d_in[11];
    const float* lin0b  = (const float*)d_in[12];
    const float* lin1W  = (const float*)d_in[13];
    const float* lin1b  = (const float*)d_in[14];
    float* out = (float*)d_out;

    const int* src = ei;
    const int* dst = ei + EDGES;

    float* buf0  = (float*)d_ws;            // x' storage between convs
    float* buf1  = buf0 + NELEM;            // agg, then h2
    float* buf2  = buf1 + NELEM;            // h1
    float* hcat  = buf2 + NELEM;            // [1024,128]
    float* stats = hcat + GRAPHS * HCATC;   // 64
    float* ss    = stats + 64;              // 64

    const int ZB_HCAT  = (GRAPHS * HCATC + 255) / 256;  // 512
    const int ZB_ELEM  = NELEM / 256;                   // 25000
    const int EB       = (EDGES * 32) / 256;            // 400000
    const int GB       = (NODES + 127) / 128;           // 1563

    // init concat pooling buffer and pool raw x into column block 0
    k_zero<<<ZB_HCAT, 256, 0, stream>>>(hcat, GRAPHS * HCATC);
    k_pool<<<ZB_ELEM, 256, 0, stream>>>(x, batch, hcat, 0);

    for (int i = 0; i < 3; i++) {
        const float* xin = (i == 0) ? x : buf0;
        // agg = segment_sum(x[src], dst)
        k_zero<<<ZB_ELEM, 256, 0, stream>>>(buf1, NELEM);
        k_edge_agg<<<EB, 256, 0, stream>>>(xin, src, dst, buf1);
        // h1 = (x + agg) @ W1 + b1  (+ fused inner-BN stats)
        k_zero<<<1, 64, 0, stream>>>(stats, 64);
        k_gemm32<<<GB, 256, 0, stream>>>(xin, buf1, ss, W1s + i * 1024,
                                         b1s + i * 32, buf2, stats, 0);
        k_finalize<<<1, 32, 0, stream>>>(stats, bng + i * 32, bnb + i * 32, ss);
        // h2 = relu(bn(h1)) @ W2 + b2  (+ fused outer-BN stats)
        k_zero<<<1, 64, 0, stream>>>(stats, 64);
        k_gemm32<<<GB, 256, 0, stream>>>(buf2, nullptr, ss, W2s + i * 1024,
                                         b2s + i * 32, buf1, stats, 1);
        k_finalize<<<1, 32, 0, stream>>>(stats, obng + i * 32, obnb + i * 32, ss);
        // x' = relu(bn(h2)); pool into concat block i+1
        k_bn_relu_pool<<<ZB_ELEM, 256, 0, stream>>>(buf1, ss, batch, buf0, hcat,
                                                    (i + 1) * 32);
    }

    k_readout<<<(GRAPHS + 255) / 256, 256, 0, stream>>>(hcat, lin0W, lin0b,
                                                        lin1W, lin1b, out);
}